// MMCL_11914239279314
// MI455X (gfx1250) — compile-verified
//
#include <hip/hip_runtime.h>
#include <cstdint>
#include <cstddef>

#define BLOCK 256
#define MAX_N 13056   // LDS key capacity (multiple of 4); N = 13000 for this problem

// GCC-vector int4 matching the builtin's expected pointee type (V4i).
typedef int v4i_ __attribute__((vector_size(16)));

// ---- order-preserving float <-> uint key mapping (on raw bits) -------------
__device__ __forceinline__ unsigned int f2key_bits(unsigned int u) {
  // negative floats: flip all bits; non-negative: set sign bit.
  return (u & 0x80000000u) ? ~u : (u | 0x80000000u);
}
__device__ __forceinline__ float key2f(unsigned int k) {
  unsigned int u = (k & 0x80000000u) ? (k & 0x7FFFFFFFu) : ~k;
  return __uint_as_float(u);
}

// ---------------------------------------------------------------------------
// One workgroup per row. Row (52 KB) lives entirely in LDS (320 KB/WGP on
// CDNA5). Staged in via the gfx1250 async global->LDS DMA path (ASYNCcnt +
// s_wait_asynccnt); fallback is vectorized b128 loads + global_prefetch.
// Then: 4-pass radix-select of the k-th largest key, one summation sweep.
// ---------------------------------------------------------------------------
__global__ void __launch_bounds__(BLOCK)
mmcl_row_kernel(const float* __restrict__ inputs,
                const int*   __restrict__ targets,
                float*       __restrict__ row_loss,
                int n, int k, float delta)
{
  __shared__ __align__(16) unsigned int keys[MAX_N];
  __shared__ unsigned int hist[256];
  __shared__ unsigned int scal[2];
  __shared__ float        fpart[BLOCK];

  const int row = blockIdx.x;
  const int tid = threadIdx.x;
  const int tgt = targets[row];
  const float* rowp = inputs + (size_t)row * (size_t)n;
  const int nv4 = n >> 2;

  // Prefetch the positive logit's line early (used only at the very end).
  if (tid == 0) __builtin_prefetch((const void*)(rowp + tgt), 0, 3);

  // ---------------- Phase A: row -> LDS keys --------------------------------
#if __has_builtin(__builtin_amdgcn_global_load_async_to_lds_b128)
  // Async DMA: copy raw bits into LDS (ASYNCcnt path), convert in place.
  for (int v = tid; v < nv4; v += BLOCK) {
    __builtin_amdgcn_global_load_async_to_lds_b128(
        (__attribute__((address_space(1))) v4i_*)(rowp + (v << 2)),
        (__attribute__((address_space(3))) v4i_*)(&keys[v << 2]),
        0, 0);
  }
  for (int i = (nv4 << 2) + tid; i < n; i += BLOCK)       // scalar tail
    keys[i] = __float_as_uint(rowp[i]);
#if __has_builtin(__builtin_amdgcn_s_wait_asynccnt)
  __builtin_amdgcn_s_wait_asynccnt(0);
#else
  asm volatile("s_wait_asynccnt 0" ::: "memory");
#endif
  __syncthreads();
  {
    uint4* k4 = (uint4*)keys;
    for (int v = tid; v < nv4; v += BLOCK) {
      uint4 u = k4[v];
      const int i = v << 2;
      u.x = (i + 0 == tgt) ? 0u : f2key_bits(u.x);
      u.y = (i + 1 == tgt) ? 0u : f2key_bits(u.y);
      u.z = (i + 2 == tgt) ? 0u : f2key_bits(u.z);
      u.w = (i + 3 == tgt) ? 0u : f2key_bits(u.w);
      k4[v] = u;
    }
    for (int i = (nv4 << 2) + tid; i < n; i += BLOCK)
      keys[i] = (i == tgt) ? 0u : f2key_bits(keys[i]);
  }
#else
  // Fallback: vectorized b128 loads with software prefetch (global_prefetch_b8).
  {
    const float4* r4 = (const float4*)rowp;
    for (int v = tid; v < nv4; v += BLOCK) {
      __builtin_prefetch((const void*)(r4 + v + 4 * BLOCK), 0, 0);
      float4 f = r4[v];
      const int i = v << 2;
      keys[i + 0] = (i + 0 == tgt) ? 0u : f2key_bits(__float_as_uint(f.x));
      keys[i + 1] = (i + 1 == tgt) ? 0u : f2key_bits(__float_as_uint(f.y));
      keys[i + 2] = (i + 2 == tgt) ? 0u : f2key_bits(__float_as_uint(f.z));
      keys[i + 3] = (i + 3 == tgt) ? 0u : f2key_bits(__float_as_uint(f.w));
    }
    for (int i = (nv4 << 2) + tid; i < n; i += BLOCK)
      keys[i] = (i == tgt) ? 0u : f2key_bits(__float_as_uint(rowp[i]));
  }
#endif
  __syncthreads();

  // ---------------- Phase B: 4-pass radix select of k-th largest key --------
  unsigned int prefix = 0u, pmask = 0u;
  int need = k;                        // elements still needed from candidates
  for (int pass = 0; pass < 4; ++pass) {
    const int shift = 24 - (pass << 3);
    hist[tid] = 0u;                    // BLOCK == 256 == #bins
    __syncthreads();
    for (int i = tid; i < n; i += BLOCK) {
      const unsigned int key = keys[i];
      if ((key & pmask) == prefix)
        atomicAdd(&hist[(key >> shift) & 0xFFu], 1u);   // ds_add_u32
    }
    __syncthreads();
    if (tid == 0) {
      unsigned int acc = 0u;
      int b = 255;
      for (; b > 0; --b) {
        const unsigned int h = hist[b];
        if (acc + h >= (unsigned int)need) break;
        acc += h;
      }
      scal[0] = prefix | ((unsigned int)b << shift);
      scal[1] = (unsigned int)need - acc;
    }
    __syncthreads();
    prefix = scal[0];
    need   = (int)scal[1];
    pmask |= 0xFFu << shift;
  }
  // prefix == exact key of the k-th largest element (T); `need` = how many
  // elements equal to T belong to the top-k (tie handling).

  // ---------------- Phase C: sum (1 + x)^2 over the top-k -------------------
  const unsigned int Tkey = prefix;
  float s = 0.0f;
  for (int i = tid; i < n; i += BLOCK) {
    const unsigned int key = keys[i];
    if (key > Tkey) {
      const float t = 1.0f + key2f(key);
      s = fmaf(t, t, s);
    }
  }
  fpart[tid] = s;
  __syncthreads();
  for (int w = BLOCK >> 1; w > 0; w >>= 1) {     // fixed-order tree reduce
    if (tid < w) fpart[tid] += fpart[tid + w];
    __syncthreads();
  }
  if (tid == 0) {
    const float T  = key2f(Tkey);
    const float t1 = 1.0f + T;
    const float negsum = fpart[0] + (float)need * (t1 * t1);
    const float pos = rowp[tgt];
    const float d   = 1.0f - pos;
    row_loss[row] = delta * d * d + negsum / (float)k;
  }
}

// ---------------------------------------------------------------------------
// Deterministic final reduction: fixed strided partials + fixed tree order.
// ---------------------------------------------------------------------------
__global__ void __launch_bounds__(BLOCK)
mmcl_reduce_kernel(const float* __restrict__ row_loss, float* __restrict__ out, int m)
{
  __shared__ float buf[BLOCK];
  const int tid = threadIdx.x;
  float s = 0.0f;
  for (int i = tid; i < m; i += BLOCK) s += row_loss[i];
  buf[tid] = s;
  __syncthreads();
  for (int w = BLOCK >> 1; w > 0; w >>= 1) {
    if (tid < w) buf[tid] += buf[tid + w];
    __syncthreads();
  }
  if (tid == 0) out[0] = buf[0] / (float)m;
}

extern "C" void kernel_launch(void* const* d_in, const int* in_sizes, int n_in,
                              void* d_out, int out_size, void* d_ws, size_t ws_size,
                              hipStream_t stream)
{
  // setup_inputs order: inputs[M*N] f32, targets_[M] (unused), targets[M], GT_MC[M] (unused)
  const float* inputs  = (const float*)d_in[0];
  const int*   targets = (const int*)  d_in[2];

  const int m = in_sizes[2];                 // 4096
  const int n = in_sizes[0] / m;             // 13000
  const int k = (int)(0.01 * (double)(n - 1));   // int(R * num_neg) = 129

  float* row_loss = (float*)d_ws;            // m floats of scratch

  mmcl_row_kernel<<<dim3(m), dim3(BLOCK), 0, stream>>>(
      inputs, targets, row_loss, n, k, 5.0f /* DELTA */);
  mmcl_reduce_kernel<<<dim3(1), dim3(BLOCK), 0, stream>>>(
      row_loss, (float*)d_out, m);
}